// CTCLossLayer_15556371546964
// MI455X (gfx1250) — compile-verified
//
#include <hip/hip_runtime.h>

// Problem constants (match reference)
#define B_     256
#define T_     512
#define C_     128
#define L_     64
#define S_     (2 * L_ + 1)   // 129
#define BLANK_ (C_ - 1)       // 127
#define NEG_   (-1.0e30f)
#define EPS_   (1e-7f)
#define LN2_   (0.69314718055994531f)

#define K_     5              // states per lane: 32*5 = 160 >= 129
#define NBUF_  8              // LDS ring slots (power of 2), 8*512B = 4KB
#define PF_    6              // async prefetch depth (NBUF_ >= PF_ + 2)

// CDNA5 async global->LDS path (ASYNCcnt-tracked), guarded per-toolchain.
#if defined(__AMDGCN__) && defined(__gfx1250__) && \
    __has_builtin(__builtin_amdgcn_global_load_async_to_lds_b128) && \
    __has_builtin(__builtin_amdgcn_s_wait_asynccnt)
#define USE_ASYNC 1
#else
#define USE_ASYNC 0
#endif

#if USE_ASYNC
typedef int v4i_t __attribute__((ext_vector_type(4)));
typedef __attribute__((address_space(1))) v4i_t* g_v4i_ptr;
typedef __attribute__((address_space(3))) v4i_t* l_v4i_ptr;
#define WAITA(n) do { __builtin_amdgcn_s_wait_asynccnt(n); \
                      asm volatile("" ::: "memory"); } while (0)
#else
#define WAITA(n) do { } while (0)
#endif

// Prefetch one 512B probability row (C_=128 floats) into the LDS ring.
// One b128 per lane: 32 lanes * 16B = 512B.
__device__ __forceinline__ void issue_row(const float* __restrict__ gbase,
                                          float* lds_rows, int t, int lane) {
  const float* g = gbase + (size_t)t * C_ + lane * 4;
  float*       l = lds_rows + (t & (NBUF_ - 1)) * C_ + lane * 4;
#if USE_ASYNC
  __builtin_amdgcn_global_load_async_to_lds_b128(
      (g_v4i_ptr)g, (l_v4i_ptr)l, 0, 0);
#else
  const float4 v = *(const float4*)g;
  *(float4*)l = v;
#endif
}

// One wave32 per batch element. alpha lives in 5 registers per lane
// (state s = lane + 32*k, log2 domain). Shifts s-1/s-2 via wave shuffles.
__global__ __launch_bounds__(32) void ctc_loss_fwd(
    const float* __restrict__ y_pred,     // (B,T,C)
    const int*   __restrict__ labels,     // (B,L)
    const int*   __restrict__ input_len,  // (B,1)
    const int*   __restrict__ label_len,  // (B,1)
    float*       __restrict__ out) {      // (B,)
  __shared__ float rows[NBUF_ * C_];
  __shared__ float afin[32 * K_];

  const int b    = blockIdx.x;
  const int lane = threadIdx.x;
  const float* gbase = y_pred + (size_t)b * T_ * C_;

  // Start the async pipeline immediately: rows 0..PF_-1 (always in-bounds,
  // PF_ < T_). Overlaps with all scalar/label setup below.
#pragma unroll
  for (int i = 0; i < PF_; ++i) issue_row(gbase, rows, i, lane);

  const int ll    = label_len[b];
  const int lastT = input_len[b] - 1;
  const int* lbl  = labels + b * L_;

  // Per-state constants: extended label, validity, skip-connection mask.
  int  extk[K_];
  bool validk[K_], skipk[K_];
#pragma unroll
  for (int k = 0; k < K_; ++k) {
    const int s = lane + 32 * k;
    int e = BLANK_; bool valid = false, skip = false;
    if (s < S_) {
      if (s & 1) e = lbl[s >> 1];
      valid = (s < 2 * ll + 1);
      if (s >= 2 && e != BLANK_) {
        const int em2 = (s & 1) ? lbl[(s >> 1) - 1] : BLANK_;
        skip = (e != em2);
      }
    }
    extk[k] = e; validk[k] = valid; skipk[k] = skip;
  }

  float a[K_];

  // One time step of the banded log-sum-exp recurrence (log2 domain).
  auto step = [&](int t) {
    const float* r = rows + (t & (NBUF_ - 1)) * C_;
    float lp[K_];
#pragma unroll
    for (int k = 0; k < K_; ++k)
      lp[k] = validk[k] ? __builtin_amdgcn_logf(r[extk[k]] + EPS_) : NEG_;

    float prev31 = NEG_, prev30 = NEG_;
#pragma unroll
    for (int k = 0; k < K_; ++k) {
      const float cur = a[k];
      const float up1 = __shfl_up(cur, 1);
      const float up2 = __shfl_up(cur, 2);
      const float c31 = __shfl(cur, 31);
      const float c30 = __shfl(cur, 30);
      const float s1  = (lane == 0) ? prev31 : up1;
      float       s2  = (lane == 0) ? prev30 : ((lane == 1) ? prev31 : up2);
      s2 = skipk[k] ? s2 : NEG_;
      const float m   = fmaxf(fmaxf(cur, s1), s2);
      const float sum = __builtin_amdgcn_exp2f(cur - m) +
                        __builtin_amdgcn_exp2f(s1 - m) +
                        __builtin_amdgcn_exp2f(s2 - m);
      a[k] = m + __builtin_amdgcn_logf(sum) + lp[k];
      prev31 = c31; prev30 = c30;
    }
  };

  // ---- t = 0 (peeled): alpha0 from row 0; keep pipeline full (issue row PF_).
  WAITA(PF_ - 1);  // rows 0..PF_-1 issued; guarantees row 0 complete
  {
    const float* r = rows;  // row 0 is in slot 0
#pragma unroll
    for (int k = 0; k < K_; ++k) {
      const int s = lane + 32 * k;
      const float lp =
          validk[k] ? __builtin_amdgcn_logf(r[extk[k]] + EPS_) : NEG_;
      a[k] = (s <= 1) ? lp : NEG_;
    }
  }
  issue_row(gbase, rows, PF_, lane);  // row PF_ -> slot PF_ (row 0 slot untouched)

  // ---- Steady state: at top of iter t, rows 0..t+PF_-1 are issued
  // (count t+PF_), so wait(PF_-1) => rows 0..t complete. Issue t+PF_ (<= T_-1).
  const int tEnd = (lastT < T_ - 1 - PF_) ? lastT : (T_ - 1 - PF_);
  for (int t = 1; t <= tEnd; ++t) {
    WAITA(PF_ - 1);
    issue_row(gbase, rows, t + PF_, lane);
    step(t);
  }
  // ---- Tail: all remaining rows (..min(lastT+PF_, T_-1)) already issued.
  for (int t = tEnd + 1; t <= lastT; ++t) {
    WAITA(0);
    step(t);
  }

  // Gather the two terminal states and emit the loss (convert back by ln2).
#pragma unroll
  for (int k = 0; k < K_; ++k) afin[lane + 32 * k] = a[k];
  __syncthreads();
  if (lane == 0) {
    const float eb = afin[2 * ll];
    const float el = afin[2 * ll - 1];
    const float m  = fmaxf(eb, el);
    const float l2 = m + __builtin_amdgcn_logf(__builtin_amdgcn_exp2f(eb - m) +
                                               __builtin_amdgcn_exp2f(el - m));
    out[b] = -LN2_ * l2;
  }
}

extern "C" void kernel_launch(void* const* d_in, const int* in_sizes, int n_in,
                              void* d_out, int out_size, void* d_ws, size_t ws_size,
                              hipStream_t stream) {
  const float* y_pred    = (const float*)d_in[0];
  const int*   labels    = (const int*)d_in[1];
  const int*   input_len = (const int*)d_in[2];
  const int*   label_len = (const int*)d_in[3];
  float*       out       = (float*)d_out;
  (void)in_sizes; (void)n_in; (void)out_size; (void)d_ws; (void)ws_size;
  ctc_loss_fwd<<<dim3(B_), dim3(32), 0, stream>>>(y_pred, labels, input_len,
                                                  label_len, out);
}